// DirGNNWithGCNEncoder_2044404433335
// MI455X (gfx1250) — compile-verified
//
#include <hip/hip_runtime.h>
#include <cstdint>

// ---------------------------------------------------------------------------
// DirGNN (2-layer directional GCN) for MI455X / gfx1250.
//  - GEMMs via V_WMMA_F32_16X16X4_F32 (full fp32, matches fp32 reference).
//    Weights pre-transposed once so A and B fragments are single contiguous
//    b64 loads; each wave computes a 16x64 strip (4 WMMA accumulators) to
//    reuse A 4x; fragment loads are software-pipelined one k-step ahead so
//    the load counter never has to drain to 0 in front of a WMMA.
//  - Edge aggregation: wave-per-edge coalesced gather + global f32 atomics
//    (entire working set fits in the 192MB L2 -> L2-BW bound).
// ---------------------------------------------------------------------------

typedef __attribute__((ext_vector_type(2))) float v2f;
typedef __attribute__((ext_vector_type(8))) float v8f;

#define N_NODES 100000
#define N_EDGES 800000
#define F_INDIM 128
#define H_DIM   128
#define C_DIM   64

// ---------------------------------------------------------------- utilities
__global__ void zero_f32_kernel(float* __restrict__ p, int n) {
    int i = blockIdx.x * blockDim.x + threadIdx.x;
    if (i < n) p[i] = 0.0f;
}

__global__ void degree_kernel(const int* __restrict__ src,
                              const int* __restrict__ dst,
                              float* __restrict__ deg_in,
                              float* __restrict__ deg_out, int E) {
    int e = blockIdx.x * blockDim.x + threadIdx.x;
    if (e < E) {
        atomicAdd(&deg_in[dst[e]], 1.0f);   // in-degree (segment over dst)
        atomicAdd(&deg_out[src[e]], 1.0f);  // out-degree (segment over src)
    }
}

__global__ void rsqrt_deg_kernel(float* __restrict__ p, int n) {
    int i = blockIdx.x * blockDim.x + threadIdx.x;
    if (i < n) {
        float d = p[i];
        p[i] = (d > 0.0f) ? __frsqrt_rn(d) : 0.0f;  // deg>=1 when >0
    }
}

__global__ void relu_kernel(float* __restrict__ p, int n) {
    int i = blockIdx.x * blockDim.x + threadIdx.x;
    if (i < n) p[i] = fmaxf(p[i], 0.0f);
}

// out[c*rows + r] = in[r*cols + c]   (rows = K, cols = Hout)
__global__ void transpose_kernel(const float* __restrict__ in,
                                 float* __restrict__ out, int rows, int cols) {
    int i = blockIdx.x * blockDim.x + threadIdx.x;
    if (i < rows * cols) {
        int r = i / cols, c = i % cols;
        out[(size_t)c * rows + r] = in[i];
    }
}

// ------------------------------------------------------ fused triple GEMM
// One 16x64 output strip per wave32 (4 x 16x16 WMMA tiles sharing A).
// O0 = X*W0, O1 = X*W1, O2 = X*W2 + bias.
// X: rows x K (row major).  WT*: transposed weights, Hout x K (row major).
// rows % 16 == 0, K % 4 == 0, Hout % 64 == 0  (Hout = 128 or 64 here).
//
// VGPR layouts per CDNA5 ISA 7.12.2 (fp32, 16x16x4):
//   A (16x4):  lane L holds M = L%16, K = 2*(L/16) + {vgpr0, vgpr1}
//   B (4x16):  vgpr j, lane L holds K = j + 2*(L/16), N = L%16
//   C (16x16): vgpr j, lane L holds M = j + 8*(L/16), N = L%16
__global__ void gemm3_wmma_kernel(const float* __restrict__ X, int K,
                                  const float* __restrict__ WT0,
                                  const float* __restrict__ WT1,
                                  const float* __restrict__ WT2,
                                  const float* __restrict__ bias,  // for O2
                                  float* __restrict__ O0,
                                  float* __restrict__ O1,
                                  float* __restrict__ O2,
                                  int rows, int Hout) {
    const int lane = threadIdx.x & 31;
    const int wave = (blockIdx.x * blockDim.x + threadIdx.x) >> 5;
    const int grpPerMat  = Hout >> 6;           // 64-wide col groups / matrix
    const int grpsPerRow = 3 * grpPerMat;
    const int rowTiles   = rows >> 4;
    if (wave >= rowTiles * grpsPerRow) return;  // wave-uniform: EXEC all-1s

    const int rowTile = wave / grpsPerRow;
    const int grp     = wave % grpsPerRow;
    const int mat     = grp / grpPerMat;
    const int colGrp  = grp % grpPerMat;

    const float* __restrict__ WT = (mat == 0) ? WT0 : (mat == 1) ? WT1 : WT2;
    float*       __restrict__ O  = (mat == 0) ? O0  : (mat == 1) ? O1  : O2;

    const int lm = lane & 15;        // M (A/C) / N (B/C)
    const int kh = lane >> 4;        // lane-half: K pair (A/B) / M-half (C)
    const int rowBase = rowTile << 4;
    const int colBase = colGrp << 6;

    // Per-lane base pointers; every fragment load is one contiguous b64.
    const float* __restrict__ aRow  = X  + (size_t)(rowBase + lm)      * K + 2 * kh;
    const float* __restrict__ bRow0 = WT + (size_t)(colBase + lm)      * K + 2 * kh;
    const float* __restrict__ bRow1 = WT + (size_t)(colBase + 16 + lm) * K + 2 * kh;
    const float* __restrict__ bRow2 = WT + (size_t)(colBase + 32 + lm) * K + 2 * kh;
    const float* __restrict__ bRow3 = WT + (size_t)(colBase + 48 + lm) * K + 2 * kh;

    v8f acc0 = {}, acc1 = {}, acc2 = {}, acc3 = {};

    // --- software-pipelined K loop: fragments loaded one k-step ahead ----
    v2f a  = *(const v2f*)(aRow);
    v2f b0 = *(const v2f*)(bRow0);
    v2f b1 = *(const v2f*)(bRow1);
    v2f b2 = *(const v2f*)(bRow2);
    v2f b3 = *(const v2f*)(bRow3);

#pragma unroll 4
    for (int k = 4; k < K; k += 4) {
        v2f an  = *(const v2f*)(aRow  + k);   // issue next-iter loads first:
        v2f b0n = *(const v2f*)(bRow0 + k);   // they stay in flight while the
        v2f b1n = *(const v2f*)(bRow1 + k);   // four WMMAs below execute
        v2f b2n = *(const v2f*)(bRow2 + k);
        v2f b3n = *(const v2f*)(bRow3 + k);
        acc0 = __builtin_amdgcn_wmma_f32_16x16x4_f32(false, a, false, b0,
                                                     (short)0, acc0, false, false);
        acc1 = __builtin_amdgcn_wmma_f32_16x16x4_f32(false, a, false, b1,
                                                     (short)0, acc1, false, false);
        acc2 = __builtin_amdgcn_wmma_f32_16x16x4_f32(false, a, false, b2,
                                                     (short)0, acc2, false, false);
        acc3 = __builtin_amdgcn_wmma_f32_16x16x4_f32(false, a, false, b3,
                                                     (short)0, acc3, false, false);
        a = an; b0 = b0n; b1 = b1n; b2 = b2n; b3 = b3n;
    }
    acc0 = __builtin_amdgcn_wmma_f32_16x16x4_f32(false, a, false, b0,
                                                 (short)0, acc0, false, false);
    acc1 = __builtin_amdgcn_wmma_f32_16x16x4_f32(false, a, false, b1,
                                                 (short)0, acc1, false, false);
    acc2 = __builtin_amdgcn_wmma_f32_16x16x4_f32(false, a, false, b2,
                                                 (short)0, acc2, false, false);
    acc3 = __builtin_amdgcn_wmma_f32_16x16x4_f32(false, a, false, b3,
                                                 (short)0, acc3, false, false);

    float bv0 = 0.f, bv1 = 0.f, bv2 = 0.f, bv3 = 0.f;
    if (mat == 2 && bias != nullptr) {
        bv0 = bias[colBase + lm];
        bv1 = bias[colBase + 16 + lm];
        bv2 = bias[colBase + 32 + lm];
        bv3 = bias[colBase + 48 + lm];
    }

#pragma unroll
    for (int j = 0; j < 8; ++j) {
        size_t o = (size_t)(rowBase + j + 8 * kh) * Hout + colBase + lm;
        O[o]      = acc0[j] + bv0;
        O[o + 16] = acc1[j] + bv1;
        O[o + 32] = acc2[j] + bv2;
        O[o + 48] = acc3[j] + bv3;
    }
}

// ----------------------------------------------- directional edge scatter
// For each edge e (one wave32 per edge):
//   acc[dst] += 0.5 * dinv_in[src]*dinv_in[dst]   * h_in [src]
//   acc[src] += 0.5 * dinv_out[src]*dinv_out[dst] * h_out[dst]
__global__ void edge_scatter_kernel(const int* __restrict__ src,
                                    const int* __restrict__ dst,
                                    const float* __restrict__ dinv_in,
                                    const float* __restrict__ dinv_out,
                                    const float* __restrict__ h_in,
                                    const float* __restrict__ h_out,
                                    float* __restrict__ acc,
                                    int E, int F) {
    const int lane = threadIdx.x & 31;
    const int e = (blockIdx.x * blockDim.x + threadIdx.x) >> 5;
    if (e >= E) return;

    const int s = src[e];
    const int d = dst[e];
    const float ni = 0.5f * dinv_in[s]  * dinv_in[d];
    const float no = 0.5f * dinv_out[s] * dinv_out[d];

    const float* __restrict__ gi = h_in  + (size_t)s * F;
    const float* __restrict__ go = h_out + (size_t)d * F;
    float* __restrict__ ad = acc + (size_t)d * F;
    float* __restrict__ as = acc + (size_t)s * F;

    for (int f = lane; f < F; f += 32) {      // coalesced across the wave
        atomicAdd(&ad[f], ni * gi[f]);
        atomicAdd(&as[f], no * go[f]);
    }
}

// ---------------------------------------------------------------------------
extern "C" void kernel_launch(void* const* d_in, const int* in_sizes, int n_in,
                              void* d_out, int out_size, void* d_ws, size_t ws_size,
                              hipStream_t stream) {
    const float* x       = (const float*)d_in[0];
    const int*   eidx    = (const int*)d_in[1];     // (2, E): row0 = src, row1 = dst
    const float* W1_in   = (const float*)d_in[2];
    const float* W1_out  = (const float*)d_in[3];
    const float* W1_root = (const float*)d_in[4];
    const float* b1      = (const float*)d_in[5];
    const float* W2_in   = (const float*)d_in[6];
    const float* W2_out  = (const float*)d_in[7];
    const float* W2_root = (const float*)d_in[8];
    const float* b2      = (const float*)d_in[9];
    float* out = (float*)d_out;

    const int* src = eidx;
    const int* dst = eidx + N_EDGES;

    // workspace carve-out (floats)
    float* ws       = (float*)d_ws;
    float* dinv_in  = ws;                                   // N
    float* dinv_out = dinv_in  + N_NODES;                   // N
    float* bufA     = dinv_out + N_NODES;                   // N*128 (h_in)
    float* bufB     = bufA + (size_t)N_NODES * H_DIM;       // N*128 (h_out)
    float* bufAcc   = bufB + (size_t)N_NODES * H_DIM;       // N*128 (layer-1 acc)
    float* wt       = bufAcc + (size_t)N_NODES * H_DIM;     // transposed weights
    float* wt1_in   = wt;                                   // H x F_IN
    float* wt1_out  = wt1_in  + F_INDIM * H_DIM;
    float* wt1_root = wt1_out + F_INDIM * H_DIM;
    float* wt2_in   = wt1_root + F_INDIM * H_DIM;           // C x H
    float* wt2_out  = wt2_in  + H_DIM * C_DIM;
    float* wt2_root = wt2_out + H_DIM * C_DIM;

    const int TB = 256;                 // 8 waves / block
    const int wavesPerBlock = TB / 32;

    // 0. transpose the six (tiny) weight matrices -> WT is Hout x K row-major
    {
        int n1 = F_INDIM * H_DIM, n2 = H_DIM * C_DIM;
        transpose_kernel<<<(n1 + TB - 1) / TB, TB, 0, stream>>>(W1_in,   wt1_in,   F_INDIM, H_DIM);
        transpose_kernel<<<(n1 + TB - 1) / TB, TB, 0, stream>>>(W1_out,  wt1_out,  F_INDIM, H_DIM);
        transpose_kernel<<<(n1 + TB - 1) / TB, TB, 0, stream>>>(W1_root, wt1_root, F_INDIM, H_DIM);
        transpose_kernel<<<(n2 + TB - 1) / TB, TB, 0, stream>>>(W2_in,   wt2_in,   H_DIM, C_DIM);
        transpose_kernel<<<(n2 + TB - 1) / TB, TB, 0, stream>>>(W2_out,  wt2_out,  H_DIM, C_DIM);
        transpose_kernel<<<(n2 + TB - 1) / TB, TB, 0, stream>>>(W2_root, wt2_root, H_DIM, C_DIM);
    }

    // 1. degrees -> dinv
    zero_f32_kernel<<<(2 * N_NODES + TB - 1) / TB, TB, 0, stream>>>(dinv_in, 2 * N_NODES);
    degree_kernel<<<(N_EDGES + TB - 1) / TB, TB, 0, stream>>>(src, dst, dinv_in, dinv_out, N_EDGES);
    rsqrt_deg_kernel<<<(2 * N_NODES + TB - 1) / TB, TB, 0, stream>>>(dinv_in, 2 * N_NODES);

    // 2. layer 1 GEMMs: bufA = x*W1_in, bufB = x*W1_out, bufAcc = x*W1_root + b1
    {
        int strips = (N_NODES / 16) * (3 * (H_DIM / 64));
        int blocks = (strips + wavesPerBlock - 1) / wavesPerBlock;
        gemm3_wmma_kernel<<<blocks, TB, 0, stream>>>(
            x, F_INDIM, wt1_in, wt1_out, wt1_root, b1,
            bufA, bufB, bufAcc, N_NODES, H_DIM);
    }

    // 3. layer 1 edge aggregation into bufAcc, then ReLU
    {
        int blocks = (N_EDGES + wavesPerBlock - 1) / wavesPerBlock;
        edge_scatter_kernel<<<blocks, TB, 0, stream>>>(
            src, dst, dinv_in, dinv_out, bufA, bufB, bufAcc, N_EDGES, H_DIM);
    }
    relu_kernel<<<((size_t)N_NODES * H_DIM + TB - 1) / TB, TB, 0, stream>>>(
        bufAcc, N_NODES * H_DIM);

    // 4. layer 2 GEMMs: bufA = h*W2_in, bufB = h*W2_out, out = h*W2_root + b2
    {
        int strips = (N_NODES / 16) * (3 * (C_DIM / 64));
        int blocks = (strips + wavesPerBlock - 1) / wavesPerBlock;
        gemm3_wmma_kernel<<<blocks, TB, 0, stream>>>(
            bufAcc, H_DIM, wt2_in, wt2_out, wt2_root, b2,
            bufA, bufB, out, N_NODES, C_DIM);
    }

    // 5. layer 2 edge aggregation directly into d_out
    {
        int blocks = (N_EDGES + wavesPerBlock - 1) / wavesPerBlock;
        edge_scatter_kernel<<<blocks, TB, 0, stream>>>(
            src, dst, dinv_in, dinv_out, bufA, bufB, out, N_EDGES, C_DIM);
    }
}